// my_model_27925877358757
// MI455X (gfx1250) — compile-verified
//
#include <hip/hip_runtime.h>
#include <math.h>

// ---------------------------------------------------------------------------
// MI455X (gfx1250) implementation of the reference model forward pass.
//
// Roofline: dominated by two (512 x 11000) @ (11000 x 2500) fp32 GEMMs
// (~28 GFLOP each, ~245 FLOP/byte) -> compute-bound -> matrix pipe.
// V_WMMA_F32_16X16X4_F32 (native fp32 WMMA, wave32) keeps reference numerics.
//
// GEMM design:
//   * one wave owns a 16x32 C block (two 16x16 WMMA tiles sharing the A frag)
//   * hot K-loop has NO bounds checks: out-of-range M/N lanes are clamped to
//     in-bounds addresses; their garbage only reaches C entries the store
//     epilogue never writes (A row m -> C row m, B col n -> C col n).
//     K is exact: main loop over K&~3 + one guarded tail step.
//   * manual software pipeline: next K-step fragments are loaded before the
//     current WMMAs issue, so waits become partial (latency hiding), not
//     s_wait_loadcnt 0 in front of every WMMA.
//   * opm weights (110 MB) are L2-resident (192 MB) -> direct global streaming.
// ---------------------------------------------------------------------------

#define BATCH 512
#define IC 50
#define TP 110
#define HID 100
#define NC 2

typedef __attribute__((ext_vector_type(2))) float v2f;
typedef __attribute__((ext_vector_type(8))) float v8f;

__device__ __forceinline__ float actf(float x, int act) {
  if (act == 1) return x > 0.f ? x : 0.f;          // relu
  if (act == 2) return x / (1.f + __expf(-x));     // silu
  return x;
}

__device__ __forceinline__ v8f wmma4(v2f a, v2f b, v8f c) {
  return __builtin_amdgcn_wmma_f32_16x16x4_f32(false, a, false, b,
                                               (short)0, c, false, false);
}

// ---------------------------------------------------------------------------
// C[b] = act(alpha * A[b] @ B[b](^T) + bias + Res[b])
// A: (M,K) row-major, B: (K,N) row-major (or (N,K) if transB), C: (M,N).
// One wave per 16x32 block of C. blockDim = (32,4).
// ---------------------------------------------------------------------------
__global__ void k_gemm(const float* __restrict__ A, const float* __restrict__ Bm,
                       const float* __restrict__ bias, const float* __restrict__ Res,
                       float* __restrict__ C,
                       int M, int N, int K,
                       long long sA, long long sB, long long sC, long long sRes,
                       int transB, float alpha, int act) {
  const int lane  = threadIdx.x;                              // 0..31
  const int tn2   = blockIdx.x * blockDim.y + threadIdx.y;    // 16x32 block col
  const int tileM = blockIdx.y;
  const int b     = blockIdx.z;
  if (tn2 * 32 >= N) return;

  const float* Ab = A  + (long long)b * sA;
  const float* Bb = Bm + (long long)b * sB;
  float*       Cb = C  + (long long)b * sC;

  const int half = lane >> 4;        // 0 or 1
  const int l    = lane & 15;
  const int mrow = tileM * 16 + l;   // A row this lane supplies
  const int n0   = tn2 * 32 + l;     // tile-0 column
  const int n1   = n0 + 16;          // tile-1 column
  // Clamp to in-bounds addresses; garbage only reaches unstored C entries.
  const int mrC = (mrow < M) ? mrow : (M - 1);
  const int n0C = (n0   < N) ? n0   : (N - 1);
  const int n1C = (n1   < N) ? n1   : (N - 1);

  v8f acc0 = {0.f, 0.f, 0.f, 0.f, 0.f, 0.f, 0.f, 0.f};
  v8f acc1 = {0.f, 0.f, 0.f, 0.f, 0.f, 0.f, 0.f, 0.f};

  // Lane-private streaming pointers (K advances by 4 per WMMA).
  const float* pa = Ab + (long long)mrC * K + half * 2;
  const float* pb0;
  const float* pb1;
  long long bd, bstep;
  if (!transB) {
    pb0 = Bb + (long long)(half * 2) * N + n0C;
    pb1 = Bb + (long long)(half * 2) * N + n1C;
    bd = N; bstep = (long long)4 * N;
  } else {
    pb0 = Bb + (long long)n0C * K + half * 2;
    pb1 = Bb + (long long)n1C * K + half * 2;
    bd = 1; bstep = 4;
  }

  const int K4 = K & ~3;
  if (K4 >= 4) {
    // prologue load
    v2f a, b0, b1;
    a.x  = pa[0];   a.y  = pa[1];
    b0.x = pb0[0];  b0.y = pb0[bd];
    b1.x = pb1[0];  b1.y = pb1[bd];
#pragma unroll 2
    for (int k0 = 4; k0 < K4; k0 += 4) {
      pa += 4; pb0 += bstep; pb1 += bstep;
      // issue next-step loads BEFORE consuming current fragments
      v2f an, bn0, bn1;
      an.x  = pa[0];   an.y  = pa[1];
      bn0.x = pb0[0];  bn0.y = pb0[bd];
      bn1.x = pb1[0];  bn1.y = pb1[bd];
      acc0 = wmma4(a, b0, acc0);
      acc1 = wmma4(a, b1, acc1);
      a = an; b0 = bn0; b1 = bn1;
    }
    acc0 = wmma4(a, b0, acc0);
    acc1 = wmma4(a, b1, acc1);
  }

  if (K4 < K) {  // K tail (runs once; guarded, zero-filled)
    const int ka = K4 + half * 2;
    v2f a, b0, b1;
    a.x = (ka     < K) ? Ab[(long long)mrC * K + ka]     : 0.f;
    a.y = (ka + 1 < K) ? Ab[(long long)mrC * K + ka + 1] : 0.f;
    if (!transB) {
      b0.x = (ka     < K) ? Bb[(long long)(ka)     * N + n0C] : 0.f;
      b0.y = (ka + 1 < K) ? Bb[(long long)(ka + 1) * N + n0C] : 0.f;
      b1.x = (ka     < K) ? Bb[(long long)(ka)     * N + n1C] : 0.f;
      b1.y = (ka + 1 < K) ? Bb[(long long)(ka + 1) * N + n1C] : 0.f;
    } else {
      b0.x = (ka     < K) ? Bb[(long long)n0C * K + ka]     : 0.f;
      b0.y = (ka + 1 < K) ? Bb[(long long)n0C * K + ka + 1] : 0.f;
      b1.x = (ka     < K) ? Bb[(long long)n1C * K + ka]     : 0.f;
      b1.y = (ka + 1 < K) ? Bb[(long long)n1C * K + ka + 1] : 0.f;
    }
    acc0 = wmma4(a, b0, acc0);
    acc1 = wmma4(a, b1, acc1);
  }

  // Store epilogue: the only place bounds are enforced.
  const bool n0ok = (n0 < N);
  const bool n1ok = (n1 < N);
  const float bias0 = (bias && n0ok) ? bias[n0] : 0.f;
  const float bias1 = (bias && n1ok) ? bias[n1] : 0.f;
  const float* Rb = Res ? (Res + (long long)b * sRes) : (const float*)0;
#pragma unroll
  for (int r = 0; r < 8; ++r) {
    const int m = tileM * 16 + r + half * 8;
    if (m >= M) continue;
    const long long ro = (long long)m * N;
    if (n0ok) {
      float v = acc0[r] * alpha + bias0;
      if (Rb) v += Rb[ro + n0];
      Cb[ro + n0] = actf(v, act);
    }
    if (n1ok) {
      float v = acc1[r] * alpha + bias1;
      if (Rb) v += Rb[ro + n1];
      Cb[ro + n1] = actf(v, act);
    }
  }
}

// ---------------------------------------------------------------------------
// Row softmax, in place. One 128-thread block per row of length L.
// ---------------------------------------------------------------------------
__global__ void k_softmax(float* __restrict__ sm, int L) {
  float* p = sm + (long long)blockIdx.x * L;
  __shared__ float red[128];
  const int t = threadIdx.x;
  float m = -3.0e38f;
  for (int i = t; i < L; i += 128) m = fmaxf(m, p[i]);
  red[t] = m; __syncthreads();
  for (int o = 64; o; o >>= 1) { if (t < o) red[t] = fmaxf(red[t], red[t + o]); __syncthreads(); }
  m = red[0]; __syncthreads();
  float s = 0.f;
  for (int i = t; i < L; i += 128) { float e = __expf(p[i] - m); p[i] = e; s += e; }
  red[t] = s; __syncthreads();
  for (int o = 64; o; o >>= 1) { if (t < o) red[t] += red[t + o]; __syncthreads(); }
  const float inv = 1.f / red[0];
  for (int i = t; i < L; i += 128) p[i] *= inv;
}

// ---------------------------------------------------------------------------
// Training-mode BatchNorm: stats per channel over (B, L). x layout (B, C, L).
// ---------------------------------------------------------------------------
__global__ void k_bn_stats(const float* __restrict__ x, float* __restrict__ mean,
                           float* __restrict__ invstd, int Bv, int C, int L, float eps) {
  const int c = blockIdx.x;
  __shared__ float s1[256], s2[256];
  const int t = threadIdx.x;
  const long long n = (long long)Bv * L;
  float sum = 0.f, sq = 0.f;
  for (long long i = t; i < n; i += 256) {
    const long long b = i / L;
    const int l = (int)(i % L);
    const float v = x[((long long)b * C + c) * L + l];
    sum += v; sq += v * v;
  }
  s1[t] = sum; s2[t] = sq; __syncthreads();
  for (int o = 128; o; o >>= 1) {
    if (t < o) { s1[t] += s1[t + o]; s2[t] += s2[t + o]; }
    __syncthreads();
  }
  if (t == 0) {
    const float mu  = s1[0] / (float)n;
    const float var = s2[0] / (float)n - mu * mu;
    mean[c]   = mu;
    invstd[c] = rsqrtf(var + eps);
  }
}

__global__ void k_bn_apply(const float* __restrict__ x, float* __restrict__ y,
                           const float* __restrict__ mean, const float* __restrict__ invstd,
                           const float* __restrict__ g, const float* __restrict__ bt,
                           int C, int L, long long total, int act) {
  const long long i = (long long)blockIdx.x * blockDim.x + threadIdx.x;
  if (i >= total) return;
  const int c = (int)((i / L) % C);
  y[i] = actf((x[i] - mean[c]) * invstd[c] * g[c] + bt[c], act);
}

// (B,R,C) -> (B,C,R) transpose with optional activation
__global__ void k_transpose(const float* __restrict__ x, float* __restrict__ y,
                            int Bv, int R, int Cc, int act) {
  const long long total = (long long)Bv * R * Cc;
  const long long i = (long long)blockIdx.x * blockDim.x + threadIdx.x;
  if (i >= total) return;
  const int c = (int)(i % Cc);
  const long long tq = i / Cc;
  const int r = (int)(tq % R);
  const int b = (int)(tq / R);
  y[((long long)b * Cc + c) * R + r] = actf(x[i], act);
}

__global__ void k_add(const float* __restrict__ a, const float* __restrict__ b,
                      float* __restrict__ y, long long total, int act) {
  const long long i = (long long)blockIdx.x * blockDim.x + threadIdx.x;
  if (i < total) y[i] = actf(a[i] + b[i], act);
}

__global__ void k_act(const float* __restrict__ x, float* __restrict__ y,
                      long long total, int act) {
  const long long i = (long long)blockIdx.x * blockDim.x + threadIdx.x;
  if (i < total) y[i] = actf(x[i], act);
}

// AvgPool1d(k=3,s=1,p=1, count_include_pad) and MaxPool1d(k=3,s=1,p=1,-inf pad)
__global__ void k_pool3(const float* __restrict__ ap, const float* __restrict__ mp,
                        float* __restrict__ apo, float* __restrict__ mpo,
                        int L, long long rows) {
  const long long i = (long long)blockIdx.x * blockDim.x + threadIdx.x;
  if (i >= rows * L) return;
  const long long r = i / L;
  const int j = (int)(i % L);
  const float* a = ap + r * L;
  const float* m = mp + r * L;
  float s = a[j];
  if (j > 0)     s += a[j - 1];
  if (j < L - 1) s += a[j + 1];
  apo[i] = s * (1.f / 3.f);
  float mx = m[j];
  if (j > 0)     mx = fmaxf(mx, m[j - 1]);
  if (j < L - 1) mx = fmaxf(mx, m[j + 1]);
  mpo[i] = mx;
}

// cosine similarity over dim=1 of (B, Ic, Lc) pairs
__global__ void k_csim(const float* __restrict__ ap, const float* __restrict__ mp,
                       float* __restrict__ cs, int Bv, int Icn, int Lc) {
  const int idx = blockIdx.x * blockDim.x + threadIdx.x;
  if (idx >= Bv * Lc) return;
  const int b = idx / Lc, j = idx % Lc;
  const float* a = ap + ((long long)b * Icn) * Lc + j;
  const float* m = mp + ((long long)b * Icn) * Lc + j;
  float num = 0.f, na = 0.f, nb = 0.f;
  for (int i = 0; i < Icn; ++i) {
    const float x = a[(long long)i * Lc];
    const float y = m[(long long)i * Lc];
    num += x * y; na += x * x; nb += y * y;
  }
  const float den = fmaxf(sqrtf(na), 1e-8f) * fmaxf(sqrtf(nb), 1e-8f);
  cs[idx] = num / den;
}

// glob[b,i,j] = cs[b,j] * fnc[b,i,j]   (fnc: (B,IC,IC))
__global__ void k_globscale(const float* __restrict__ fnc, const float* __restrict__ cs,
                            float* __restrict__ y, int Bv) {
  const long long total = (long long)Bv * IC * IC;
  const long long i = (long long)blockIdx.x * blockDim.x + threadIdx.x;
  if (i >= total) return;
  const int j = (int)(i % IC);
  const int b = (int)(i / (IC * IC));
  y[i] = fnc[i] * cs[b * IC + j];
}

// out[b*outStride + outOff + i] = mean_t x[b*sB + i*sI + t*sT]
__global__ void k_meanstride(const float* __restrict__ x, float* __restrict__ out,
                             int Bv, int I, int T,
                             long long sB, long long sI, long long sT,
                             int outStride, int outOff) {
  const int idx = blockIdx.x * blockDim.x + threadIdx.x;
  if (idx >= Bv * I) return;
  const int b = idx / I, i = idx % I;
  const float* p = x + (long long)b * sB + (long long)i * sI;
  float s = 0.f;
  for (int t = 0; t < T; ++t) s += p[(long long)t * sT];
  out[b * outStride + outOff + i] = s / (float)T;
}

// concat along last axis: y (rows, C1+C2) from a (rows,C1), b (rows,C2)
__global__ void k_concat_last(const float* __restrict__ a, const float* __restrict__ bs,
                              float* __restrict__ y, long long rows, int C1, int C2) {
  const int Ct = C1 + C2;
  const long long total = rows * Ct;
  const long long i = (long long)blockIdx.x * blockDim.x + threadIdx.x;
  if (i >= total) return;
  const long long r = i / Ct;
  const int c = (int)(i % Ct);
  y[i] = (c < C1) ? a[r * C1 + c] : bs[r * C2 + (c - C1)];
}

// fm2T[b,i,j] = (j<IC) ? f[b,j,i] : g[b,j-IC,i]   with f,g (B,IC,IC), out (B,IC,2*IC)
__global__ void k_fncmap2T(const float* __restrict__ f, const float* __restrict__ g,
                           float* __restrict__ y, int Bv) {
  const long long total = (long long)Bv * IC * (2 * IC);
  const long long i = (long long)blockIdx.x * blockDim.x + threadIdx.x;
  if (i >= total) return;
  const int j = (int)(i % (2 * IC));
  const long long tq = i / (2 * IC);
  const int r = (int)(tq % IC);
  const int b = (int)(tq / IC);
  y[i] = (j < IC) ? f[((long long)b * IC + j) * IC + r]
                  : g[((long long)b * IC + (j - IC)) * IC + r];
}

// opv[r,h] = o1[r,h] * mean_h(o2[r,:])
__global__ void k_opm_mix(const float* __restrict__ o1, const float* __restrict__ o2,
                          float* __restrict__ y, int rows, int H) {
  const int r = blockIdx.x * blockDim.x + threadIdx.x;
  if (r >= rows) return;
  const float* p = o2 + (long long)r * H;
  float s = 0.f;
  for (int h = 0; h < H; ++h) s += p[h];
  s /= (float)H;
  const float* q = o1 + (long long)r * H;
  float* o = y + (long long)r * H;
  for (int h = 0; h < H; ++h) o[h] = q[h] * s;
}

// ---------------------------------------------------------------------------
// Host-side helpers
// ---------------------------------------------------------------------------
static void gemm(hipStream_t st, const float* A, const float* Bm, const float* bias,
                 const float* Res, float* C, int M, int N, int K,
                 int batch, long long sA, long long sB, long long sC, long long sRes,
                 int transB, float alpha, int act) {
  dim3 blk(32, 4, 1);
  const unsigned pairs = (unsigned)((N + 31) / 32);
  dim3 grd((pairs + 3) / 4, (unsigned)((M + 15) / 16), (unsigned)batch);
  k_gemm<<<grd, blk, 0, st>>>(A, Bm, bias, Res, C, M, N, K, sA, sB, sC, sRes,
                              transB, alpha, act);
}

static void lin(hipStream_t st, const float* A, const float* W, const float* b,
                const float* Res, float* C, int M, int N, int K, int act) {
  gemm(st, A, W, b, Res, C, M, N, K, 1, 0, 0, 0, 0, 0, 1.f, act);
}

#define EW(total) ((unsigned)(((total) + 255) / 256)), 256

static void run_bn(hipStream_t st, const float* x, float* y, const float* g,
                   const float* b, int C, int L, float* mean, float* invstd, int act) {
  k_bn_stats<<<(unsigned)C, 256, 0, st>>>(x, mean, invstd, BATCH, C, L, 1e-5f);
  const long long total = (long long)BATCH * C * L;
  k_bn_apply<<<EW(total), 0, st>>>(x, y, mean, invstd, g, b, C, L, total, act);
}

// single-head self-attention with residual; W = (4,D,D) [q,k,v,o], b = (4,D)
static void run_attn(hipStream_t st, const float* x, int S, int D,
                     const float* W, const float* bias, float* out, int act,
                     float* q, float* k, float* v, float* s) {
  const float scale = 1.0f / sqrtf((float)D);
  lin(st, x, W,                        bias,         nullptr, q, BATCH * S, D, D, 0);
  lin(st, x, W + (long long)D * D,     bias + D,     nullptr, k, BATCH * S, D, D, 0);
  lin(st, x, W + (long long)2 * D * D, bias + 2 * D, nullptr, v, BATCH * S, D, D, 0);
  // s[b] = q[b] @ k[b]^T * scale
  gemm(st, q, k, nullptr, nullptr, s, S, S, D, BATCH,
       (long long)S * D, (long long)S * D, (long long)S * S, 0, 1, scale, 0);
  k_softmax<<<(unsigned)(BATCH * S), 128, 0, st>>>(s, S);
  // o[b] = softmax(s)[b] @ v[b]  (reuse q as o)
  gemm(st, s, v, nullptr, nullptr, q, S, D, S, BATCH,
       (long long)S * S, (long long)S * D, (long long)S * D, 0, 0, 1.f, 0);
  // out = act(o @ Wo + bo + x)
  lin(st, q, W + (long long)3 * D * D, bias + 3 * D, x, out, BATCH * S, D, D, act);
}

static void run_opm(hipStream_t st, const float* x /*(B,TP,IC)*/,
                    const float* W11, const float* b11, const float* W12, const float* b12,
                    const float* W2, const float* b2, const float* gamma, const float* beta,
                    float* out /*(B,IC*IC)*/, float* o1, float* o2, float* opv,
                    float* mean, float* invstd) {
  lin(st, x, W11, b11, nullptr, o1, BATCH * TP, HID, IC, 0);
  lin(st, x, W12, b12, nullptr, o2, BATCH * TP, HID, IC, 0);
  k_opm_mix<<<EW(BATCH * TP), 0, st>>>(o1, o2, opv, BATCH * TP, HID);
  lin(st, opv, W2, b2, nullptr, out, BATCH, IC * IC, HID * TP, 0);
  run_bn(st, out, out, gamma, beta, IC * IC, 1, mean, invstd, 0);
}

// ---------------------------------------------------------------------------
// Input index map. Assumes JAX-canonical tree flattening of setup_inputs():
// top-level keys sorted ('fnc','params','tcs'); nested dict keys sorted.
// ---------------------------------------------------------------------------
enum {
  IN_FNC = 0, IN_LR_W, IN_LR_b,
  IN_bm1_b, IN_bm1_g, IN_bm2_b, IN_bm2_g, IN_bm3_b, IN_bm3_g,
  IN_bm4_b, IN_bm4_g, IN_bm5_b, IN_bm5_g, IN_bm6_b, IN_bm6_g,
  IN_bm_b, IN_bm_g,
  IN_f2t1_W, IN_f2t1_b, IN_f2t2_W, IN_f2t2_b,
  IN_fcAtt1_W, IN_fcAtt1_b, IN_fcAtt2_W, IN_fcAtt2_b,
  IN_fc_W, IN_fc_b,
  IN_glb_W1, IN_glb_W2, IN_glb_b1, IN_glb_b2,
  IN_o1_W11, IN_o1_W12, IN_o1_W2, IN_o1_b11, IN_o1_b12, IN_o1_b2, IN_o1_beta, IN_o1_gamma,
  IN_o2_W11, IN_o2_W12, IN_o2_W2, IN_o2_b11, IN_o2_b12, IN_o2_b2, IN_o2_beta, IN_o2_gamma,
  IN_sAtt1_W, IN_sAtt1_b, IN_sAtt2_W, IN_sAtt2_b,
  IN_tAtt1_W, IN_tAtt1_b, IN_tAtt2_W, IN_tAtt2_b,
  IN_TCS, IN_COUNT
};

struct Bump { char* p; };
static float* balloc(Bump& w, size_t nfloats) {
  float* r = (float*)w.p;
  w.p += ((nfloats * sizeof(float)) + 255) & ~(size_t)255;
  return r;
}

extern "C" void kernel_launch(void* const* d_in, const int* in_sizes, int n_in,
                              void* d_out, int out_size, void* d_ws, size_t ws_size,
                              hipStream_t stream) {
  (void)in_sizes; (void)n_in; (void)out_size; (void)ws_size;
  const float* P[IN_COUNT];
  for (int i = 0; i < IN_COUNT; ++i) P[i] = (const float*)d_in[i];
  const float* fnc = P[IN_FNC];
  const float* tcs = P[IN_TCS];

  float* out_head = (float*)d_out;                  // (B, NC)
  float* bm_out   = (float*)d_out + BATCH * NC;     // (B, IC)  (post-silu bm)

  Bump w{(char*)d_ws};

  // Shared attention scratch (max: B*TP*IC for q/k/v/o, B*TP*TP for scores)
  float* q  = balloc(w, (size_t)BATCH * TP * IC);
  float* kb = balloc(w, (size_t)BATCH * TP * IC);
  float* vb = balloc(w, (size_t)BATCH * TP * IC);
  float* sb = balloc(w, (size_t)BATCH * TP * TP);
  float* mean   = balloc(w, IC * IC);
  float* invstd = balloc(w, IC * IC);

  // ----- Mask (glob) -----
  float* ap  = balloc(w, (size_t)BATCH * IC * IC);
  float* mp  = balloc(w, (size_t)BATCH * IC * IC);
  float* apP = balloc(w, (size_t)BATCH * IC * IC);
  float* mpP = balloc(w, (size_t)BATCH * IC * IC);
  float* cs  = balloc(w, (size_t)BATCH * IC);
  float* glob = balloc(w, (size_t)BATCH * IC * IC);
  lin(stream, fnc, P[IN_glb_W1], P[IN_glb_b1], nullptr, ap, BATCH * IC, IC, IC, 0);
  lin(stream, fnc, P[IN_glb_W2], P[IN_glb_b2], nullptr, mp, BATCH * IC, IC, IC, 0);
  k_pool3<<<EW((long long)BATCH * IC * IC), 0, stream>>>(ap, mp, apP, mpP, IC, (long long)BATCH * IC);
  k_csim<<<EW(BATCH * IC), 0, stream>>>(apP, mpP, cs, BATCH, IC, IC);
  k_globscale<<<EW((long long)BATCH * IC * IC), 0, stream>>>(fnc, cs, glob, BATCH);

  float* bmcat = balloc(w, (size_t)BATCH * 5 * IC);
  // globbm -> bmcat[:, 0:50]
  k_meanstride<<<EW(BATCH * IC), 0, stream>>>(glob, bmcat, BATCH, IC, IC,
                                              (long long)IC * IC, IC, 1, 5 * IC, 0);

  // ----- fnc branch: fcAtt1 + relu + bn1 -----
  float* fout = balloc(w, (size_t)BATCH * IC * IC);
  run_attn(stream, fnc, IC, IC, P[IN_fcAtt1_W], P[IN_fcAtt1_b], fout, 1, q, kb, vb, sb);
  run_bn(stream, fout, fout, P[IN_bm1_g], P[IN_bm1_b], IC, IC, mean, invstd, 0);

  float* fncmap1 = balloc(w, (size_t)BATCH * IC * 2 * IC);   // (B,IC,2IC) concat last axis
  k_concat_last<<<EW((long long)BATCH * IC * 2 * IC), 0, stream>>>(fout, glob, fncmap1,
                                                                   (long long)BATCH * IC, IC, IC);

  // ----- tc branch -----
  float* tc0 = balloc(w, (size_t)BATCH * TP * IC);   // silu(swap(tcs))
  k_transpose<<<EW((long long)BATCH * IC * TP), 0, stream>>>(tcs, tc0, BATCH, IC, TP, 2);

  float* tc1 = balloc(w, (size_t)BATCH * TP * IC);
  run_attn(stream, tc0, TP, IC, P[IN_sAtt1_W], P[IN_sAtt1_b], tc1, 2, q, kb, vb, sb);
  run_bn(stream, tc1, tc1, P[IN_bm2_g], P[IN_bm2_b], TP, IC, mean, invstd, 0);

  float* tc2 = balloc(w, (size_t)BATCH * IC * TP);
  k_transpose<<<EW((long long)BATCH * TP * IC), 0, stream>>>(tc1, tc2, BATCH, TP, IC, 0);

  float* tc3 = balloc(w, (size_t)BATCH * IC * TP);   // tcout after tAtt1 + bn3
  run_attn(stream, tc2, IC, TP, P[IN_tAtt1_W], P[IN_tAtt1_b], tc3, 2, q, kb, vb, sb);
  run_bn(stream, tc3, tc3, P[IN_bm3_g], P[IN_bm3_b], IC, TP, mean, invstd, 0);

  // ----- opm1 -----
  float* xop = balloc(w, (size_t)BATCH * TP * IC);
  k_transpose<<<EW((long long)BATCH * IC * TP), 0, stream>>>(tc3, xop, BATCH, IC, TP, 0);
  float* o1  = balloc(w, (size_t)BATCH * TP * HID);
  float* o2  = balloc(w, (size_t)BATCH * TP * HID);
  float* opv = balloc(w, (size_t)BATCH * TP * HID);
  float* tcmap1 = balloc(w, (size_t)BATCH * IC * IC);
  run_opm(stream, xop, P[IN_o1_W11], P[IN_o1_b11], P[IN_o1_W12], P[IN_o1_b12],
          P[IN_o1_W2], P[IN_o1_b2], P[IN_o1_gamma], P[IN_o1_beta],
          tcmap1, o1, o2, opv, mean, invstd);

  // tcout_b = tc3 + lin(fncmap1, f2t1)
  float* tcb = balloc(w, (size_t)BATCH * IC * TP);
  lin(stream, fncmap1, P[IN_f2t1_W], P[IN_f2t1_b], tc3, tcb, BATCH * IC, TP, 2 * IC, 0);

  // fout_r = fout + tcmap1 ; fncbm1 -> bmcat[:,100:150]
  float* foutr = balloc(w, (size_t)BATCH * IC * IC);
  k_add<<<EW((long long)BATCH * IC * IC), 0, stream>>>(fout, tcmap1, foutr,
                                                       (long long)BATCH * IC * IC, 0);
  k_meanstride<<<EW(BATCH * IC), 0, stream>>>(foutr, bmcat, BATCH, IC, IC,
                                              (long long)IC * IC, IC, 1, 5 * IC, 2 * IC);

  // fcAtt2 + silu + bn4
  float* fout4 = balloc(w, (size_t)BATCH * IC * IC);
  run_attn(stream, foutr, IC, IC, P[IN_fcAtt2_W], P[IN_fcAtt2_b], fout4, 2, q, kb, vb, sb);
  run_bn(stream, fout4, fout4, P[IN_bm4_g], P[IN_bm4_b], IC, IC, mean, invstd, 0);

  // tAtt2 + silu + bn5 on tcb (B,IC,TP)
  float* tc4 = balloc(w, (size_t)BATCH * IC * TP);
  run_attn(stream, tcb, IC, TP, P[IN_tAtt2_W], P[IN_tAtt2_b], tc4, 2, q, kb, vb, sb);
  run_bn(stream, tc4, tc4, P[IN_bm5_g], P[IN_bm5_b], IC, TP, mean, invstd, 0);

  // swap -> sAtt2 + silu + bn6
  float* tc5T = balloc(w, (size_t)BATCH * TP * IC);
  k_transpose<<<EW((long long)BATCH * IC * TP), 0, stream>>>(tc4, tc5T, BATCH, IC, TP, 0);
  float* tc6 = balloc(w, (size_t)BATCH * TP * IC);
  run_attn(stream, tc5T, TP, IC, P[IN_sAtt2_W], P[IN_sAtt2_b], tc6, 2, q, kb, vb, sb);
  run_bn(stream, tc6, tc6, P[IN_bm6_g], P[IN_bm6_b], TP, IC, mean, invstd, 0);

  // tcbm1 = mean over t of tc6[b,t,i] -> bmcat[:,50:100]
  k_meanstride<<<EW(BATCH * IC), 0, stream>>>(tc6, bmcat, BATCH, IC, TP,
                                              (long long)TP * IC, 1, IC, 5 * IC, IC);

  // tc8 = silu(tc6); tcmap2 = opm(tc8)
  float* tc8 = balloc(w, (size_t)BATCH * TP * IC);
  k_act<<<EW((long long)BATCH * TP * IC), 0, stream>>>(tc6, tc8, (long long)BATCH * TP * IC, 2);
  float* tcmap2 = balloc(w, (size_t)BATCH * IC * IC);
  run_opm(stream, tc8, P[IN_o2_W11], P[IN_o2_b11], P[IN_o2_W12], P[IN_o2_b12],
          P[IN_o2_W2], P[IN_o2_b2], P[IN_o2_gamma], P[IN_o2_beta],
          tcmap2, o1, o2, opv, mean, invstd);

  // tcout_c = swap(tc8) + lin(swap(fncmap2), f2t2)
  float* tc8T = balloc(w, (size_t)BATCH * IC * TP);
  k_transpose<<<EW((long long)BATCH * TP * IC), 0, stream>>>(tc8, tc8T, BATCH, TP, IC, 0);
  float* fm2T = balloc(w, (size_t)BATCH * IC * 2 * IC);
  k_fncmap2T<<<EW((long long)BATCH * IC * 2 * IC), 0, stream>>>(fout4, glob, fm2T, BATCH);
  float* tcc = balloc(w, (size_t)BATCH * IC * TP);
  lin(stream, fm2T, P[IN_f2t2_W], P[IN_f2t2_b], tc8T, tcc, BATCH * IC, TP, 2 * IC, 0);

  // fout5 = fout4 + tcmap2
  float* fout5 = balloc(w, (size_t)BATCH * IC * IC);
  k_add<<<EW((long long)BATCH * IC * IC), 0, stream>>>(fout4, tcmap2, fout5,
                                                       (long long)BATCH * IC * IC, 0);

  // tcbm2 -> bmcat[:,150:200]; fncbm2 -> bmcat[:,200:250]
  k_meanstride<<<EW(BATCH * IC), 0, stream>>>(tcc, bmcat, BATCH, IC, TP,
                                              (long long)IC * TP, TP, 1, 5 * IC, 3 * IC);
  k_meanstride<<<EW(BATCH * IC), 0, stream>>>(fout5, bmcat, BATCH, IC, IC,
                                              (long long)IC * IC, IC, 1, 5 * IC, 4 * IC);

  // bm = silu(bmcat); bm = bn(fc(bm)); bm = silu -> bm_out; out = LR(bm)
  k_act<<<EW((long long)BATCH * 5 * IC), 0, stream>>>(bmcat, bmcat, (long long)BATCH * 5 * IC, 2);
  float* bml = balloc(w, (size_t)BATCH * IC);
  lin(stream, bmcat, P[IN_fc_W], P[IN_fc_b], nullptr, bml, BATCH, IC, 5 * IC, 0);
  run_bn(stream, bml, bm_out, P[IN_bm_g], P[IN_bm_b], IC, 1, mean, invstd, 2);
  lin(stream, bm_out, P[IN_LR_W], P[IN_LR_b], nullptr, out_head, BATCH, NC, IC, 0);
}